// SinkhornFromScores_10359461117953
// MI455X (gfx1250) — compile-verified
//
#include <hip/hip_runtime.h>

// Sinkhorn-Knopp, log domain, B=8, n=2048, TAU=1, 20 iterations.
//
// Rank-1 reformulation: la == a - u_i - v_j is invariant, so we iterate on
// potentials u,v (8*2048 f32 each) and only ever READ the 134MB matrix:
//   u_i = LSE_j(a_ij - v_j);  v_j = LSE_i(a_ij - u_i);  P = exp(a - u - v)
// Traffic: 20 iters * 2 matrix reads + final read/write ~= 5.6 GB, and the
// matrix is L2-resident (134MB < 192MB L2), so sweeps run at L2 bandwidth.
//
// Row-pass exp-sums are computed with V_WMMA_F32_16X16X4_F32 (f32 precision):
// A = exp values (every slot filled), B = ones  =>  sum(all D slots) ==
// 16 * sum(all A slots), independent of the lane/VGPR layout details.

#define NN 2048
#define BB 8
#define NITERS 20
#define CCHUNKS 16
#define CROWS (NN / CCHUNKS) /* 128 rows per column-pass chunk */

typedef __attribute__((ext_vector_type(2))) float v2f;
typedef __attribute__((ext_vector_type(8))) float v8f;

__device__ __forceinline__ float waveReduceMax(float m) {
#pragma unroll
  for (int off = 16; off >= 1; off >>= 1)
    m = fmaxf(m, __shfl_xor(m, off, 32));
  return m;
}

__device__ __forceinline__ float waveReduceSum(float s) {
#pragma unroll
  for (int off = 16; off >= 1; off >>= 1)
    s += __shfl_xor(s, off, 32);
  return s;
}

__global__ __launch_bounds__(256) void zero_vec_kernel(float* __restrict__ p) {
  p[blockIdx.x * 256 + threadIdx.x] = 0.0f;
}

// One wave (32 lanes) per row. 8 waves / block -> 8 rows / block.
// Loads the whole row into registers (64 f32/lane, coalesced float4),
// max-trick LSE with the exp-sum accumulated via WMMA f32 16x16x4.
__global__ __launch_bounds__(256) void row_lse_kernel(const float* __restrict__ a,
                                                      const float* __restrict__ v,
                                                      float* __restrict__ u) {
  const int lane = threadIdx.x & 31;
  const int wave = threadIdx.x >> 5;
  const long long row = (long long)blockIdx.x * 8 + wave;  // 0 .. BB*NN-1
  const int bIdx = (int)(row >> 11);                       // row / NN
  const float4* __restrict__ arow = (const float4*)(a + row * NN);
  const float4* __restrict__ vrow = (const float4*)(v + ((long long)bIdx << 11));

  float t[64];
  float m = -3.402823466e+38f;
#pragma unroll
  for (int k = 0; k < 16; ++k) {
    float4 av = arow[lane + 32 * k];  // fully coalesced 128-bit loads
    float4 vv = vrow[lane + 32 * k];
    float t0 = av.x - vv.x;  // TAU == 1 (a already = log_alpha / TAU)
    float t1 = av.y - vv.y;
    float t2 = av.z - vv.z;
    float t3 = av.w - vv.w;
    t[4 * k + 0] = t0;
    t[4 * k + 1] = t1;
    t[4 * k + 2] = t2;
    t[4 * k + 3] = t3;
    m = fmaxf(m, fmaxf(fmaxf(t0, t1), fmaxf(t2, t3)));
  }
  m = waveReduceMax(m);  // row max (max-trick)

  // Accumulate sum(exp(t - m)) with WMMA: D += A x Ones. Two accumulators
  // break the WMMA->WMMA accumulation dependency chain.
  v8f d0 = {};
  v8f d1 = {};
  v2f ones;
  ones.x = 1.0f;
  ones.y = 1.0f;
#pragma unroll
  for (int c = 0; c < 16; ++c) {
    v2f a0, a1;
    a0.x = __expf(t[4 * c + 0] - m);
    a0.y = __expf(t[4 * c + 1] - m);
    a1.x = __expf(t[4 * c + 2] - m);
    a1.y = __expf(t[4 * c + 3] - m);
    d0 = __builtin_amdgcn_wmma_f32_16x16x4_f32(false, a0, false, ones,
                                               (short)0, d0, false, false);
    d1 = __builtin_amdgcn_wmma_f32_16x16x4_f32(false, a1, false, ones,
                                               (short)0, d1, false, false);
  }
  v8f d = d0 + d1;
  // sum of all 256 D slots == 16 * sum of all A slots (B == ones), layout-free
  float s = d[0] + d[1] + d[2] + d[3] + d[4] + d[5] + d[6] + d[7];
  s = waveReduceSum(s) * 0.0625f;

  if (lane == 0) u[row] = m + __logf(s);
}

// Column pass, stage 1: each thread owns one column over a 128-row chunk,
// online (max, sum) logsumexp with perfectly coalesced loads.
__global__ __launch_bounds__(256) void col_partial_kernel(const float* __restrict__ a,
                                                          const float* __restrict__ u,
                                                          float* __restrict__ mPart,
                                                          float* __restrict__ sPart) {
  const int tiles = NN / 256;  // 8 column tiles per batch
  const int bIdx = blockIdx.x / tiles;
  const int j = (blockIdx.x % tiles) * 256 + threadIdx.x;
  const int chunk = blockIdx.y;

  const float* __restrict__ ab =
      a + ((long long)bIdx * NN + (long long)chunk * CROWS) * NN + j;
  const float* __restrict__ ub = u + bIdx * NN + chunk * CROWS;

  float m = -3.402823466e+38f;
  float s = 0.0f;
  for (int i = 0; i < CROWS; ++i) {
    float t = ab[(long long)i * NN] - ub[i];  // ub[i]: uniform broadcast load
    float mn = fmaxf(m, t);
    s = s * __expf(m - mn) + __expf(t - mn);
    m = mn;
  }
  const int idx = (bIdx * CCHUNKS + chunk) * NN + j;
  mPart[idx] = m;
  sPart[idx] = s;
}

// Column pass, stage 2: merge the 16 (m,s) partials per column -> v_j.
__global__ __launch_bounds__(256) void col_combine_kernel(const float* __restrict__ mPart,
                                                          const float* __restrict__ sPart,
                                                          float* __restrict__ v) {
  const int g = blockIdx.x * 256 + threadIdx.x;  // 0 .. BB*NN-1
  const int b = g >> 11;
  const int j = g & (NN - 1);
  const float* __restrict__ mp = mPart + (long long)b * CCHUNKS * NN + j;
  const float* __restrict__ sp = sPart + (long long)b * CCHUNKS * NN + j;
  float M = mp[0];
#pragma unroll
  for (int c = 1; c < CCHUNKS; ++c) M = fmaxf(M, mp[(long long)c * NN]);
  float S = 0.0f;
#pragma unroll
  for (int c = 0; c < CCHUNKS; ++c)
    S += sp[(long long)c * NN] * __expf(mp[(long long)c * NN] - M);
  v[g] = M + __logf(S);
}

// P = exp(a - u_i - v_j), vectorized float4.
__global__ __launch_bounds__(256) void final_exp_kernel(const float* __restrict__ a,
                                                        const float* __restrict__ u,
                                                        const float* __restrict__ v,
                                                        float* __restrict__ out) {
  const long long g = (long long)blockIdx.x * 256 + threadIdx.x;  // float4 index
  const long long e = g * 4;
  const int j = (int)(e & (NN - 1));
  const long long row = e >> 11;  // b*NN + i
  const int b = (int)(row >> 11);
  const float ui = u[row];
  const float* __restrict__ vb = v + ((long long)b << 11);
  float4 av = ((const float4*)a)[g];
  float4 o;
  o.x = __expf(av.x - ui - vb[j + 0]);
  o.y = __expf(av.y - ui - vb[j + 1]);
  o.z = __expf(av.z - ui - vb[j + 2]);
  o.w = __expf(av.w - ui - vb[j + 3]);
  ((float4*)out)[g] = o;
}

extern "C" void kernel_launch(void* const* d_in, const int* in_sizes, int n_in,
                              void* d_out, int out_size, void* d_ws, size_t ws_size,
                              hipStream_t stream) {
  (void)in_sizes; (void)n_in; (void)out_size; (void)ws_size;
  const float* a = (const float*)d_in[0];  // [B, n, n] f32
  float* out = (float*)d_out;              // [B, n, n] f32

  char* ws = (char*)d_ws;
  float* v = (float*)ws;                                // BB*NN f32 (64 KB)
  float* u = (float*)(ws + 64 * 1024);                  // BB*NN f32 (64 KB)
  float* mPart = (float*)(ws + 128 * 1024);             // BB*16*NN f32 (1 MB)
  float* sPart = (float*)(ws + 128 * 1024 + (1 << 20)); // BB*16*NN f32 (1 MB)

  // v = 0 every call (d_ws is poisoned; must be deterministic).
  zero_vec_kernel<<<(BB * NN) / 256, 256, 0, stream>>>(v);

  for (int it = 0; it < NITERS; ++it) {
    // u_i = LSE_j(a_ij - v_j)
    row_lse_kernel<<<(BB * NN) / 8, 256, 0, stream>>>(a, v, u);
    // v_j = LSE_i(a_ij - u_i)
    col_partial_kernel<<<dim3(BB * (NN / 256), CCHUNKS), 256, 0, stream>>>(a, u, mPart, sPart);
    col_combine_kernel<<<(BB * NN) / 256, 256, 0, stream>>>(mPart, sPart, v);
  }

  const long long totalVec4 = (long long)BB * NN * NN / 4;  // 8.39M float4
  final_exp_kernel<<<(int)(totalVec4 / 256), 256, 0, stream>>>(a, u, v, out);
}